// CrossNetLayer_17514876633094
// MI455X (gfx1250) — compile-verified
//
#include <hip/hip_runtime.h>
#include <stdint.h>

typedef float v2f __attribute__((ext_vector_type(2)));
typedef float v4f __attribute__((ext_vector_type(4)));
typedef float v8f __attribute__((ext_vector_type(8)));
typedef int   v4i __attribute__((ext_vector_type(4)));

#define B_ROWS 1024
#define D_DIM  1024
#define H_DIM  1024
#define NDEPTH 4

#define KC 32
#define A_STRIDE 36   // A tile [m][k]: 32 + 4 pad, 16B-aligned rows
#define B_STRIDE 38   // B tile [n][k] (transposed): even stride -> b64-aligned pairs

#define AS1 __attribute__((address_space(1)))
#define AS3 __attribute__((address_space(3)))

#if defined(__gfx1250__) && __has_builtin(__builtin_amdgcn_global_load_async_to_lds_b128)
#define USE_ASYNC_LDS 1
#else
#define USE_ASYNC_LDS 0
#endif

#if USE_ASYNC_LDS && __has_builtin(__builtin_amdgcn_global_load_async_to_lds_b32)
#define USE_ASYNC_B32 1
#else
#define USE_ASYNC_B32 0
#endif

__device__ __forceinline__ AS1 v4i* gcast(const void* p) {
    return (AS1 v4i*)(uintptr_t)p;                 // inttoptr keeps full 64-bit VA
}
__device__ __forceinline__ AS3 v4i* lcast(const void* p) {
    return (AS3 v4i*)(uint32_t)(uintptr_t)p;       // generic shared VA truncates to LDS offset
}

__device__ __forceinline__ void lds_copy16(const void* g, void* l) {
#if USE_ASYNC_LDS
    __builtin_amdgcn_global_load_async_to_lds_b128(gcast(g), lcast(l), 0, 0);
#else
    *(v4f*)l = *(const v4f*)g;
#endif
}

__device__ __forceinline__ void lds_copy4(const void* g, void* l) {
#if USE_ASYNC_B32
    __builtin_amdgcn_global_load_async_to_lds_b32(
        (AS1 int*)(uintptr_t)g, (AS3 int*)(uint32_t)(uintptr_t)l, 0, 0);
#else
    *(float*)l = *(const float*)g;
#endif
}

__device__ __forceinline__ void wait_async_copies() {
#if USE_ASYNC_LDS
#if __has_builtin(__builtin_amdgcn_s_wait_asynccnt)
    __builtin_amdgcn_s_wait_asynccnt(0);
#else
    asm volatile("s_wait_asynccnt 0x0" ::: "memory");
#endif
#endif
}

// Stage one K-chunk:
//   A tile 64x32 -> LDS [m][k]        : 2x async B128 per thread
//   B tile 32x32 -> LDS [n][k] (T)    : 4x async B32 per thread (transposed fill)
__device__ __forceinline__ void stage_chunk(const float* __restrict__ X,
                                            const float* __restrict__ W,
                                            int M0, int N0, int k0, int tid,
                                            float* aL, float* bT)
{
    int q   = tid;                       // A float4 slot 0..255
    int row = q >> 3;
    int c4  = (q & 7) << 2;
    lds_copy16(X + (size_t)(M0 + row) * D_DIM + k0 + c4,
               aL + row * A_STRIDE + c4);

    q   = tid + 256;                     // A float4 slot 256..511
    row = q >> 3;
    c4  = (q & 7) << 2;
    lds_copy16(X + (size_t)(M0 + row) * D_DIM + k0 + c4,
               aL + row * A_STRIDE + c4);

    const int kb = tid >> 3;             // B: this thread owns (k0+kb, N0+nb..nb+3)
    const int nb = (tid & 7) << 2;
    const float* gB = W + (size_t)(k0 + kb) * H_DIM + N0 + nb;
    #pragma unroll
    for (int i = 0; i < 4; ++i)
        lds_copy4(gB + i, bT + (nb + i) * B_STRIDE + kb);
}

// ---------------------------------------------------------------------------
// Kernel 1: h = x @ W_enc + b_enc   (f32 WMMA 16x16x4)
// 8 wave32s (4M x 2N), WG tile 64x32, K chunks of 32 double-buffered through
// LDS with async global->LDS copies overlapping the WMMA stream.
// ---------------------------------------------------------------------------
__global__ __launch_bounds__(256)
void gemm_enc_kernel(const float* __restrict__ X,
                     const float* __restrict__ W,
                     const float* __restrict__ bias,
                     float* __restrict__ Hout)
{
    __shared__ float a_lds[2][64 * A_STRIDE];
    __shared__ float b_lds[2][32 * B_STRIDE];   // transposed: [n][k]

    const int tid  = threadIdx.x;
    const int lane = tid & 31;
    const int wave = tid >> 5;
    const int wm   = wave & 3;
    const int wn   = wave >> 2;
    const int lo   = lane & 15;
    const int half = lane >> 4;

    const int M0 = blockIdx.y * 64;
    const int N0 = blockIdx.x * 32;

    v8f acc = {};
    int buf = 0;

    // prologue: fill buffer 0
    stage_chunk(X, W, M0, N0, 0, tid, a_lds[0], b_lds[0]);
    wait_async_copies();
    __syncthreads();

    for (int k0 = 0; k0 < D_DIM; k0 += KC) {
        const int nxt = buf ^ 1;
        if (k0 + KC < D_DIM)   // prefetch next chunk into the other buffer
            stage_chunk(X, W, M0, N0, k0 + KC, tid, a_lds[nxt], b_lds[nxt]);

        const float* aL = a_lds[buf];
        const float* bT = b_lds[buf];
        const float* aRow = aL + (wm * 16 + lo) * A_STRIDE;   // A[m][*]
        const float* bRow = bT + (wn * 16 + lo) * B_STRIDE;   // B^T[n][*]

        // A frag (16x4 f32): lanes 0-15 -> K={0,1}, lanes 16-31 -> K={2,3}
        // B frag (4x16 f32): VGPR v holds row K=v (lanes 0-15) / K=v+2 (16-31)
        // ka = kk + 2*half is even -> both v2f loads are 8B-aligned ds_load_b64.
        #pragma unroll
        for (int kk = 0; kk < KC; kk += 4) {
            const int ka = kk + 2 * half;
            v2f a = *(const v2f*)(aRow + ka);
            v2f b = *(const v2f*)(bRow + ka);
            acc = __builtin_amdgcn_wmma_f32_16x16x4_f32(
                /*neg_a=*/false, a, /*neg_b=*/false, b,
                /*c_mod=*/(short)0, acc, /*reuse_a=*/false, /*reuse_b=*/false);
        }

        wait_async_copies();   // next chunk landed in LDS (this wave's copies)
        __syncthreads();       // all waves done reading buf + their copies done
        buf = nxt;
    }

    // epilogue: fuse bias, write h (= x0 for the CrossNet stage)
    const int col = N0 + wn * 16 + lo;
    const float bv = bias[col];
    #pragma unroll
    for (int v = 0; v < 8; ++v) {
        const int row = M0 + wm * 16 + v + 8 * half;   // C/D layout: M = v + 8*half
        Hout[(size_t)row * H_DIM + col] = acc[v] + bv;
    }
}

// ---------------------------------------------------------------------------
// Kernel 2: CrossNet, one WG per row; row lives in registers across all 4
// layers.  s = dot(x_l, ws[l]);  x_l += x0 * s + bs[l]
// ---------------------------------------------------------------------------
__global__ __launch_bounds__(256)
void crossnet_kernel(const float* __restrict__ Hbuf,
                     const float* __restrict__ wsv,
                     const float* __restrict__ bsv,
                     float* __restrict__ out)
{
    __shared__ float red[256];

    const int row = blockIdx.x;
    const int tid = threadIdx.x;

    float x0[4], xl[4];
    #pragma unroll
    for (int k = 0; k < 4; ++k) {
        x0[k] = Hbuf[(size_t)row * H_DIM + tid + k * 256];
        xl[k] = x0[k];
    }

    for (int l = 0; l < NDEPTH; ++l) {
        float p = 0.f;
        #pragma unroll
        for (int k = 0; k < 4; ++k)
            p += xl[k] * wsv[l * H_DIM + tid + k * 256];
        red[tid] = p;
        __syncthreads();
        #pragma unroll
        for (int off = 128; off > 0; off >>= 1) {
            if (tid < off) red[tid] += red[tid + off];
            __syncthreads();
        }
        const float s = red[0];
        __syncthreads();   // red reused next layer
        #pragma unroll
        for (int k = 0; k < 4; ++k)
            xl[k] += x0[k] * s + bsv[l * H_DIM + tid + k * 256];
    }

    #pragma unroll
    for (int k = 0; k < 4; ++k)
        out[(size_t)row * H_DIM + tid + k * 256] = xl[k];
}

// ---------------------------------------------------------------------------
extern "C" void kernel_launch(void* const* d_in, const int* in_sizes, int n_in,
                              void* d_out, int out_size, void* d_ws, size_t ws_size,
                              hipStream_t stream)
{
    const float* x    = (const float*)d_in[0];   // (1024, 1024)
    const float* Wenc = (const float*)d_in[1];   // (1024, 1024)
    const float* benc = (const float*)d_in[2];   // (1024,)
    const float* wsv  = (const float*)d_in[3];   // (4, 1024, 1)
    const float* bsv  = (const float*)d_in[4];   // (4, 1024, 1)
    float* out = (float*)d_out;                  // (1024, 1024)
    float* h   = (float*)d_ws;                   // x0 scratch, 4 MB

    dim3 g1(H_DIM / 32, B_ROWS / 64);            // 32 x 16 = 512 WGs
    gemm_enc_kernel<<<g1, 256, 0, stream>>>(x, Wenc, benc, h);
    crossnet_kernel<<<B_ROWS, 256, 0, stream>>>(h, wsv, bsv, out);
}